// FPModule_80272938762724
// MI455X (gfx1250) — compile-verified
//
#include <hip/hip_runtime.h>
#include <hip/hip_bf16.h>

typedef float v2f __attribute__((ext_vector_type(2)));
typedef float v8f __attribute__((ext_vector_type(8)));

#define KNN_BLOCK 256
#define MAX_N 4096

// ---------------------------------------------------------------------------
// Kernel 1: brute-force 3-NN over LDS-cached positions + weight computation.
// Also writes the pos_skip / batch_skip components of the tuple output.
// ---------------------------------------------------------------------------
__global__ __launch_bounds__(KNN_BLOCK) void knn_kernel(
    const float* __restrict__ pos,        // [N,3]
    const float* __restrict__ pos_skip,   // [M,3]
    const long long* __restrict__ batch_skip, // [M]
    int N, int M,
    int*   __restrict__ knn_idx,          // [M,4]
    float* __restrict__ knn_w,            // [M,4] (normalized weights)
    float* __restrict__ out_pos,          // d_out + M*C_IN
    float* __restrict__ out_batch)        // d_out + M*C_IN + M*3
{
    __shared__ float s_pos[MAX_N * 3];
    const int tid = threadIdx.x;
    const int total = N * 3;
    for (int i = tid; i < total; i += KNN_BLOCK) s_pos[i] = pos[i];
    __syncthreads();

    const int m = blockIdx.x * KNN_BLOCK + tid;
    if (m >= M) return;

    const float qx = pos_skip[m * 3 + 0];
    const float qy = pos_skip[m * 3 + 1];
    const float qz = pos_skip[m * 3 + 2];

    float bd0 = 3.4e38f, bd1 = 3.4e38f, bd2 = 3.4e38f;
    int   bi0 = 0, bi1 = 0, bi2 = 0;

    for (int j = 0; j < N; ++j) {
        const float dx = qx - s_pos[j * 3 + 0];
        const float dy = qy - s_pos[j * 3 + 1];
        const float dz = qz - s_pos[j * 3 + 2];
        const float d2 = fmaf(dx, dx, fmaf(dy, dy, dz * dz));
        if (d2 < bd2) {
            if (d2 < bd1) {
                bd2 = bd1; bi2 = bi1;
                if (d2 < bd0) { bd1 = bd0; bi1 = bi0; bd0 = d2; bi0 = j; }
                else          { bd1 = d2;  bi1 = j; }
            } else { bd2 = d2; bi2 = j; }
        }
    }

    const float w0 = 1.0f / fmaxf(bd0, 1e-16f);
    const float w1 = 1.0f / fmaxf(bd1, 1e-16f);
    const float w2 = 1.0f / fmaxf(bd2, 1e-16f);
    const float inv = 1.0f / (w0 + w1 + w2);

    knn_idx[m * 4 + 0] = bi0;
    knn_idx[m * 4 + 1] = bi1;
    knn_idx[m * 4 + 2] = bi2;
    knn_idx[m * 4 + 3] = 0;
    knn_w[m * 4 + 0] = w0 * inv;
    knn_w[m * 4 + 1] = w1 * inv;
    knn_w[m * 4 + 2] = w2 * inv;
    knn_w[m * 4 + 3] = 0.0f;

    out_pos[m * 3 + 0] = qx;
    out_pos[m * 3 + 1] = qy;
    out_pos[m * 3 + 2] = qz;
    out_batch[m] = (float)batch_skip[m];
}

// ---------------------------------------------------------------------------
// Kernel 2: inverse-distance interpolation gather. 256 threads = 8 waves,
// 4 queries per block, 64 threads x float4 = 256 channels per query.
// ---------------------------------------------------------------------------
__global__ __launch_bounds__(256) void interp_kernel(
    const float* __restrict__ x,        // [N, C]
    const int*   __restrict__ knn_idx,  // [M,4]
    const float* __restrict__ knn_w,    // [M,4]
    float* __restrict__ y,              // [M, C]
    int C4, int M)                      // C/4
{
    const int q  = threadIdx.x / 64;            // query within block (0..3)
    const int c  = threadIdx.x & 63;            // channel group
    const int m  = blockIdx.x * 4 + q;
    if (m >= M || c >= C4) return;

    const int i0 = knn_idx[m * 4 + 0];
    const int i1 = knn_idx[m * 4 + 1];
    const int i2 = knn_idx[m * 4 + 2];
    const float w0 = knn_w[m * 4 + 0];
    const float w1 = knn_w[m * 4 + 1];
    const float w2 = knn_w[m * 4 + 2];

    const float4* x4 = (const float4*)x;
    const float4 a = x4[(size_t)i0 * C4 + c];
    const float4 b = x4[(size_t)i1 * C4 + c];
    const float4 d = x4[(size_t)i2 * C4 + c];

    float4 r;
    r.x = fmaf(w0, a.x, fmaf(w1, b.x, w2 * d.x));
    r.y = fmaf(w0, a.y, fmaf(w1, b.y, w2 * d.y));
    r.z = fmaf(w0, a.z, fmaf(w1, b.z, w2 * d.z));
    r.w = fmaf(w0, a.w, fmaf(w1, b.w, w2 * d.w));
    ((float4*)y)[(size_t)m * C4 + c] = r;
}

// ---------------------------------------------------------------------------
// Kernel 3/4: fp32 WMMA GEMM.  C[M,Nout] = act( [A1|A2] @ W + bias )
// Block = 128 threads (4 waves); block tile 64x64; each wave computes a
// 16x64 strip via 4 accumulators of v_wmma_f32_16x16x4_f32.
//
// fp32 WMMA fragment layout (wave32):
//   A (16x4):  lane&15 = row, lane>>4 selects K-pair {0,1}/{2,3}; v0=K, v1=K+1
//   B (4x16):  lane&15 = col, lane>>4 selects K-pair;             v0=K, v1=K+1
//   C/D:       VGPR r: lanes 0-15 -> row r, lanes 16-31 -> row r+8; lane&15 = col
//
// Split K loops (A1 then A2) + pointer-increment addressing so the inner loop
// is {1x b64 A load, 8x b32 B loads (clause), 4x WMMA} with add-only math.
// ---------------------------------------------------------------------------
__global__ __launch_bounds__(128) void gemm_wmma(
    const float* __restrict__ A1, int K1,    // [M, K1]
    const float* __restrict__ A2, int K2,    // [M, K2] or nullptr
    const float* __restrict__ W,  int ldw,   // [K1+K2, ldw]
    const float* __restrict__ bias,          // [ldw]
    float* __restrict__ C, int ldc,
    int relu)
{
    const int lane = threadIdx.x & 31;
    const int wave = threadIdx.x >> 5;
    const int m_base = blockIdx.x * 64 + wave * 16;
    const int n_base = blockIdx.y * 64;
    const int row  = lane & 15;
    const int half = lane >> 4;        // 0 or 1 -> K offset 0 / 2

    v8f acc0 = {}, acc1 = {}, acc2 = {}, acc3 = {};

    const size_t wstep = (size_t)4 * ldw;
    const float* wp  = W + (size_t)(half * 2) * ldw + n_base + row;
    const float* ap  = A1 + (size_t)(m_base + row) * K1 + half * 2;

    #pragma unroll 4
    for (int k = 0; k < K1; k += 4) {
        const v2f a = *(const v2f*)ap;
        ap += 4;
        v2f b0, b1, b2, b3;
        b0.x = wp[0];  b0.y = wp[ldw + 0];
        b1.x = wp[16]; b1.y = wp[ldw + 16];
        b2.x = wp[32]; b2.y = wp[ldw + 32];
        b3.x = wp[48]; b3.y = wp[ldw + 48];
        wp += wstep;

        acc0 = __builtin_amdgcn_wmma_f32_16x16x4_f32(false, a, false, b0, (short)0, acc0, false, false);
        acc1 = __builtin_amdgcn_wmma_f32_16x16x4_f32(false, a, false, b1, (short)0, acc1, false, false);
        acc2 = __builtin_amdgcn_wmma_f32_16x16x4_f32(false, a, false, b2, (short)0, acc2, false, false);
        acc3 = __builtin_amdgcn_wmma_f32_16x16x4_f32(false, a, false, b3, (short)0, acc3, false, false);
    }

    if (A2) {
        const float* ap2 = A2 + (size_t)(m_base + row) * K2 + half * 2;
        #pragma unroll 4
        for (int k = 0; k < K2; k += 4) {
            const v2f a = *(const v2f*)ap2;
            ap2 += 4;
            v2f b0, b1, b2, b3;
            b0.x = wp[0];  b0.y = wp[ldw + 0];
            b1.x = wp[16]; b1.y = wp[ldw + 16];
            b2.x = wp[32]; b2.y = wp[ldw + 32];
            b3.x = wp[48]; b3.y = wp[ldw + 48];
            wp += wstep;

            acc0 = __builtin_amdgcn_wmma_f32_16x16x4_f32(false, a, false, b0, (short)0, acc0, false, false);
            acc1 = __builtin_amdgcn_wmma_f32_16x16x4_f32(false, a, false, b1, (short)0, acc1, false, false);
            acc2 = __builtin_amdgcn_wmma_f32_16x16x4_f32(false, a, false, b2, (short)0, acc2, false, false);
            acc3 = __builtin_amdgcn_wmma_f32_16x16x4_f32(false, a, false, b3, (short)0, acc3, false, false);
        }
    }

    // Epilogue: bias + optional ReLU, scatter per documented D layout.
    const int col0 = n_base + row;
    const float bb0 = bias[col0];
    const float bb1 = bias[col0 + 16];
    const float bb2 = bias[col0 + 32];
    const float bb3 = bias[col0 + 48];
    #pragma unroll
    for (int r = 0; r < 8; ++r) {
        const int ro = m_base + half * 8 + r;
        float* cp = C + (size_t)ro * ldc + col0;
        float v0 = acc0[r] + bb0;
        float v1 = acc1[r] + bb1;
        float v2 = acc2[r] + bb2;
        float v3 = acc3[r] + bb3;
        if (relu) {
            v0 = fmaxf(v0, 0.0f); v1 = fmaxf(v1, 0.0f);
            v2 = fmaxf(v2, 0.0f); v3 = fmaxf(v3, 0.0f);
        }
        cp[0]  = v0;
        cp[16] = v1;
        cp[32] = v2;
        cp[48] = v3;
    }
}

// ---------------------------------------------------------------------------
// Host-side launch
// ---------------------------------------------------------------------------
extern "C" void kernel_launch(void* const* d_in, const int* in_sizes, int n_in,
                              void* d_out, int out_size, void* d_ws, size_t ws_size,
                              hipStream_t stream) {
    const float*     x          = (const float*)d_in[0];      // [N, C_IN]
    const float*     pos        = (const float*)d_in[1];      // [N, 3]
    const float*     x_skip     = (const float*)d_in[3];      // [M, C_SKIP]
    const float*     pos_skip   = (const float*)d_in[4];      // [M, 3]
    const long long* batch_skip = (const long long*)d_in[5];  // [M]
    const float*     W1         = (const float*)d_in[6];      // [C_IN+C_SKIP, H]
    const float*     b1         = (const float*)d_in[7];      // [H]
    const float*     W2         = (const float*)d_in[8];      // [H, H]
    const float*     b2         = (const float*)d_in[9];      // [H]

    const int N      = in_sizes[1] / 3;       // 4096
    const int M      = in_sizes[4] / 3;       // 16384
    const int C_IN   = in_sizes[0] / N;       // 256
    const int C_SKIP = in_sizes[3] / M;       // 128
    const int H      = in_sizes[7];           // 256

    float* outF      = (float*)d_out;
    float* out_main  = outF;                               // [M, H]
    float* out_pos   = outF + (size_t)M * H;               // [M, 3]
    float* out_batch = out_pos + (size_t)M * 3;            // [M]

    // Workspace layout
    float* y   = (float*)d_ws;                             // [M, C_IN]
    float* h   = y + (size_t)M * C_IN;                     // [M, H]
    int*   idx = (int*)(h + (size_t)M * H);                // [M, 4]
    float* w   = (float*)(idx + (size_t)M * 4);            // [M, 4]

    // 1) kNN search + weights + tuple tail outputs
    knn_kernel<<<(M + KNN_BLOCK - 1) / KNN_BLOCK, KNN_BLOCK, 0, stream>>>(
        pos, pos_skip, batch_skip, N, M, idx, w, out_pos, out_batch);

    // 2) inverse-distance interpolation gather -> y
    interp_kernel<<<(M + 3) / 4, 256, 0, stream>>>(x, idx, w, y, C_IN / 4, M);

    // 3) h = relu([y | x_skip] @ W1 + b1)   (split-K, concat never materialized)
    {
        dim3 grid(M / 64, H / 64);
        gemm_wmma<<<grid, 128, 0, stream>>>(y, C_IN, x_skip, C_SKIP,
                                            W1, H, b1, h, H, /*relu=*/1);
    }

    // 4) out = h @ W2 + b2
    {
        dim3 grid(M / 64, H / 64);
        gemm_wmma<<<grid, 128, 0, stream>>>(h, H, nullptr, 0,
                                            W2, H, b2, out_main, H, /*relu=*/0);
    }
}